// FFT256_84576495992867
// MI455X (gfx1250) — compile-verified
//
#include <hip/hip_runtime.h>

// CDNA5 / gfx1250: wave32, WMMA f32 16x16x4.
typedef __attribute__((ext_vector_type(2))) float v2f;
typedef __attribute__((ext_vector_type(8))) float v8f;

__device__ __forceinline__ v2f mk2(float a, float b) {
    v2f r; r.x = a; r.y = b; return r;
}

// Four-step FFT, N = 16 x 16, one wave32 per batch, software-pipelined:
//   Y = Xin * F16        (WMMA, A = data, B = DFT16 const)
//   Z = twiddle .* Y     (VALU, after C/D-layout -> B-layout lane permute)
//   Out = F16 * Z        (WMMA, A = DFT16 const, B = data)
// Complex matmul expanded with COS/SIN/(-SIN) constant matrices:
//   Yr = Xr*COS + Xi*SIN ;  Yi = Xi*COS + Xr*(-SIN)   (F16 = cos - i sin)
__global__ __launch_bounds__(256) void fft256_wmma_kernel(
        const float* __restrict__ x, float* __restrict__ y, int nbatch) {
    const int lane   = threadIdx.x & 31;
    const int laneLo = lane & 15;
    const int hi     = lane >> 4;

    const float STEP16  = 6.28318530717958647692f / 16.0f;
    const float STEP256 = 6.28318530717958647692f / 256.0f;

    // Batch-invariant constants, computed once per wave.
    float cosB[8], sinB[8], sinnB[8];   // DFT16 in B-operand layout (stage 1)
    float cosA[8], sinA[8], sinnA[8];   // DFT16 in A-operand layout (stage 3)
    float twc[8], tws[8];               // W256 twiddles in B-operand layout
    int   offv[8];                      // per-lane A-layout gather offsets
#pragma unroll
    for (int r = 0; r < 8; ++r) {
        const int kB = 2 * r + hi;                      // B-layout row K
        const int kA = 4 * (r >> 1) + 2 * hi + (r & 1); // A-layout col K
        offv[r] = laneLo + 16 * kA;
        const float angB = STEP16 * (float)((kB * laneLo) & 15);
        const float angA = STEP16 * (float)((kA * laneLo) & 15);
        cosB[r] = __cosf(angB); sinB[r] = __sinf(angB); sinnB[r] = -sinB[r];
        cosA[r] = __cosf(angA); sinA[r] = __sinf(angA); sinnA[r] = -sinA[r];
        const float angT = STEP256 * (float)((kB * laneLo) & 255); // n1=kB, k2=laneLo
        twc[r] = __cosf(angT); tws[r] = __sinf(angT);
    }

    const int wavesTotal = (gridDim.x * blockDim.x) >> 5;
    int b = (blockIdx.x * blockDim.x + threadIdx.x) >> 5;
    if (b >= nbatch) return;

    // Prologue: stream in the first batch (non-temporal: data is touched once).
    float ar[8], ai[8];
    {
        const float* xb = x + (size_t)b * 512;
#pragma unroll
        for (int r = 0; r < 8; ++r) {
            ar[r] = __builtin_nontemporal_load(xb + offv[r]);
            ai[r] = __builtin_nontemporal_load(xb + 256 + offv[r]);
        }
    }

    for (; b < nbatch; b += wavesTotal) {
        // Issue next batch's loads before this batch's compute so HBM latency
        // overlaps the WMMA + permute + twiddle phase.
        const int bn = b + wavesTotal;
        float nr[8], ni[8];
        if (bn < nbatch) {
            const float* xn = x + (size_t)bn * 512;
#pragma unroll
            for (int r = 0; r < 8; ++r) {
                nr[r] = __builtin_nontemporal_load(xn + offv[r]);
                ni[r] = __builtin_nontemporal_load(xn + 256 + offv[r]);
            }
        } else {
#pragma unroll
            for (int r = 0; r < 8; ++r) { nr[r] = 0.f; ni[r] = 0.f; }
        }

        // Stage 1: Y = Xin * F16  (K=16 via 4 chained 16x16x4 WMMAs).
        v8f yr = {0.f, 0.f, 0.f, 0.f, 0.f, 0.f, 0.f, 0.f};
        v8f yi = {0.f, 0.f, 0.f, 0.f, 0.f, 0.f, 0.f, 0.f};
#pragma unroll
        for (int j = 0; j < 4; ++j) {
            const v2f Ar = mk2(ar[2*j], ar[2*j+1]);
            const v2f Ai = mk2(ai[2*j], ai[2*j+1]);
            const v2f Bc = mk2(cosB[2*j], cosB[2*j+1]);
            const v2f Bs = mk2(sinB[2*j], sinB[2*j+1]);
            const v2f Bn = mk2(sinnB[2*j], sinnB[2*j+1]);
            yr = __builtin_amdgcn_wmma_f32_16x16x4_f32(false, Ar, false, Bc, (short)0, yr, false, false);
            yr = __builtin_amdgcn_wmma_f32_16x16x4_f32(false, Ai, false, Bs, (short)0, yr, false, false);
            yi = __builtin_amdgcn_wmma_f32_16x16x4_f32(false, Ai, false, Bc, (short)0, yi, false, false);
            yi = __builtin_amdgcn_wmma_f32_16x16x4_f32(false, Ar, false, Bn, (short)0, yi, false, false);
        }

        // C/D layout (row M = v + 8*hi) -> B layout (row K = 2r + hi):
        // rows 2r,2r+1 both live in source half hs = r>>2; gather across the
        // 16-lane boundary with ds_bpermute, then twiddle in B layout.
        float zrB[8], ziB[8];
#pragma unroll
        for (int r = 0; r < 8; ++r) {
            const int hs  = r >> 2;
            const int v0  = (2 * r) & 7;
            const int src = laneLo + 16 * hs;
            const float q0r = __shfl(yr[v0],     src, 32);
            const float q1r = __shfl(yr[v0 + 1], src, 32);
            const float q0i = __shfl(yi[v0],     src, 32);
            const float q1i = __shfl(yi[v0 + 1], src, 32);
            const float wr = hi ? q1r : q0r;
            const float wi = hi ? q1i : q0i;
            zrB[r] = wr * twc[r] + wi * tws[r];   // Z = W * Y, W = c - i s
            ziB[r] = wi * twc[r] - wr * tws[r];
        }

        // Stage 3: Out = F16 * Z.
        v8f outr = {0.f, 0.f, 0.f, 0.f, 0.f, 0.f, 0.f, 0.f};
        v8f outi = {0.f, 0.f, 0.f, 0.f, 0.f, 0.f, 0.f, 0.f};
#pragma unroll
        for (int j = 0; j < 4; ++j) {
            const v2f Ac = mk2(cosA[2*j], cosA[2*j+1]);
            const v2f As = mk2(sinA[2*j], sinA[2*j+1]);
            const v2f An = mk2(sinnA[2*j], sinnA[2*j+1]);
            const v2f Br = mk2(zrB[2*j], zrB[2*j+1]);
            const v2f Bi = mk2(ziB[2*j], ziB[2*j+1]);
            outr = __builtin_amdgcn_wmma_f32_16x16x4_f32(false, Ac, false, Br, (short)0, outr, false, false);
            outr = __builtin_amdgcn_wmma_f32_16x16x4_f32(false, As, false, Bi, (short)0, outr, false, false);
            outi = __builtin_amdgcn_wmma_f32_16x16x4_f32(false, Ac, false, Bi, (short)0, outi, false, false);
            outi = __builtin_amdgcn_wmma_f32_16x16x4_f32(false, An, false, Br, (short)0, outi, false, false);
        }

        // Out[k1][k2] -> linear k = 16*k1 + k2 : C/D layout stores coalesce.
        // Non-temporal: written once, never re-read by the kernel.
        float* orp = y + (size_t)b * 512;
        float* oip = orp + 256;
#pragma unroll
        for (int v = 0; v < 8; ++v) {
            const int k = 16 * (v + 8 * hi) + laneLo;
            __builtin_nontemporal_store(outr[v], orp + k);
            __builtin_nontemporal_store(outi[v], oip + k);
        }

        // Rotate double buffer.
#pragma unroll
        for (int r = 0; r < 8; ++r) { ar[r] = nr[r]; ai[r] = ni[r]; }
    }
}

extern "C" void kernel_launch(void* const* d_in, const int* in_sizes, int n_in,
                              void* d_out, int out_size, void* d_ws, size_t ws_size,
                              hipStream_t stream) {
    const float* x = (const float*)d_in[0];
    float* out = (float*)d_out;
    const int nbatch = in_sizes[0] / 512;   // 131072 for [B,2,256]

    const int threads = 256;                 // 8 wave32 per workgroup
    int waves_needed = (nbatch + 7) / 8;     // ~8 batches per wave
    int blocks = (waves_needed + 7) / 8;
    if (blocks < 1) blocks = 1;
    if (blocks > 2048) blocks = 2048;

    fft256_wmma_kernel<<<blocks, threads, 0, stream>>>(x, out, nbatch);
}